// HistLayer_16097537425431
// MI455X (gfx1250) — compile-verified
//
#include <hip/hip_runtime.h>
#include <stdint.h>

// ---------------------------------------------------------------------------
// HistLayer (differentiable hard-binning histogram) fused kernel — MI455X.
//
// Memory-bound: 50.3 MB in + 402.7 MB out => ~19.4 us floor @ 23.3 TB/s.
// Pass 1: stream input through an async global->LDS double buffer (ASYNCcnt),
//         compute all 8 bins per pixel, NT-store two_d, block-reduce partial
//         sums per (plane, chunk) into workspace.
// Pass 2: tiny deterministic reduction of 64 partials -> one_d (mean).
// No matmul in this op => WMMA intentionally unused; the CDNA5 features that
// pay off here are the async global->LDS copy path (ASYNCcnt), NT cache
// hints for the 403 MB streaming output (> 192 MB L2), and wave32 shuffles.
// ---------------------------------------------------------------------------

typedef float v4f __attribute__((ext_vector_type(4)));
typedef int   v4i __attribute__((__vector_size__(16)));   // matches builtin param

#define NBINS    8
#define HWSZ     (512 * 512)
#define THREADS  256
#define NITER    4
#define PIX_PER_BLOCK (THREADS * NITER * 4)   // 4096 pixels per block
#define NCHUNKS  (HWSZ / PIX_PER_BLOCK)       // 64 chunks per (b,c) plane
#define LOG2_1P01 0.014355292977070041f       // log2(1.01)

#if defined(__HIP_DEVICE_COMPILE__) && defined(__gfx1250__) && \
    __has_builtin(__builtin_amdgcn_global_load_async_to_lds_b128)
#define USE_ASYNC_LDS 1
#else
#define USE_ASYNC_LDS 0
#endif

__device__ __forceinline__ void stage_copy16(v4f* lds_dst, const v4f* gsrc) {
#if USE_ASYNC_LDS
  // Builtin signature (from clang diagnostic):
  //   (AS1 v4i* src, AS3 v4i* dst, imm int offset, imm int cpol)
  v4i* gflat = (v4i*)const_cast<v4f*>(gsrc);
  v4i* lflat = (v4i*)lds_dst;
  __builtin_amdgcn_global_load_async_to_lds_b128(
      (__attribute__((address_space(1))) v4i*)gflat,
      (__attribute__((address_space(3))) v4i*)lflat,
      /*offset=*/0, /*cpol=*/0);
#else
  *lds_dst = *gsrc;
#endif
}

template <int N>
__device__ __forceinline__ void wait_async() {
#if USE_ASYNC_LDS
#if __has_builtin(__builtin_amdgcn_s_wait_asynccnt)
  __builtin_amdgcn_s_wait_asynccnt(N);
#else
  asm volatile("s_wait_asynccnt %0" ::"i"(N) : "memory");
#endif
#endif
}

// ---------------------------------------------------------------------------
// Pass 1: main streaming kernel.
// grid = (NCHUNKS, planes=B*C), block = 256 (8 wave32s).
// ---------------------------------------------------------------------------
__global__ void __launch_bounds__(THREADS)
hist_main(const float* __restrict__ in,
          const float* __restrict__ centers,
          const float* __restrict__ widthp,
          float* __restrict__ two_d,
          float* __restrict__ partial) {
  const int tid   = threadIdx.x;
  const int chunk = blockIdx.x;   // 0..NCHUNKS-1
  const int plane = blockIdx.y;   // b*C + c

  const float w = widthp[0];
  float c[NBINS];
#pragma unroll
  for (int k = 0; k < NBINS; ++k) c[k] = centers[k];

  const int64_t hw0 = (int64_t)chunk * PIX_PER_BLOCK;
  const v4f* gin = (const v4f*)(in + (int64_t)plane * HWSZ + hw0);
  float* gout = two_d + (int64_t)plane * NBINS * HWSZ + hw0;

  __shared__ v4f   stage[2][THREADS];        // 8 KB double buffer
  __shared__ float ssum[THREADS / 32][NBINS];

  float acc[NBINS];
#pragma unroll
  for (int k = 0; k < NBINS; ++k) acc[k] = 0.0f;

  // Prime the pipeline.
  stage_copy16(&stage[0][tid], gin + tid);

#pragma unroll
  for (int it = 0; it < NITER; ++it) {
    if (it + 1 < NITER) {
      stage_copy16(&stage[(it + 1) & 1][tid], gin + (it + 1) * THREADS + tid);
      wait_async<1>();   // previous buffer landed in LDS
    } else {
      wait_async<0>();
    }
    // Each lane consumes only the bytes it fetched: no barrier needed.
    const v4f x = stage[it & 1][tid];
    const int64_t p0 = (int64_t)(it * THREADS + tid) * 4;  // pixel offset in chunk

#pragma unroll
    for (int k = 0; k < NBINS; ++k) {
      const float ck = c[k];
      v4f z, p, r;
      z.x = w - fabsf(x.x - ck);
      z.y = w - fabsf(x.y - ck);
      z.z = w - fabsf(x.z - ck);
      z.w = w - fabsf(x.w - ck);
      // 1.01^z = exp2(z * log2(1.01)); keep strictly > 1, else 0 (reference).
      p.x = __builtin_amdgcn_exp2f(z.x * LOG2_1P01);
      p.y = __builtin_amdgcn_exp2f(z.y * LOG2_1P01);
      p.z = __builtin_amdgcn_exp2f(z.z * LOG2_1P01);
      p.w = __builtin_amdgcn_exp2f(z.w * LOG2_1P01);
      r.x = (p.x > 1.0f) ? p.x : 0.0f;
      r.y = (p.y > 1.0f) ? p.y : 0.0f;
      r.z = (p.z > 1.0f) ? p.z : 0.0f;
      r.w = (p.w > 1.0f) ? p.w : 0.0f;
      acc[k] += (r.x + r.y) + (r.z + r.w);
      // 403 MB streaming output > 192 MB L2: non-temporal store.
      __builtin_nontemporal_store(r, (v4f*)(gout + (int64_t)k * HWSZ + p0));
    }
  }

  // ---- deterministic block reduction: wave32 shuffle tree, then LDS ----
#pragma unroll
  for (int k = 0; k < NBINS; ++k) {
    float v = acc[k];
#pragma unroll
    for (int off = 16; off > 0; off >>= 1) v += __shfl_down(v, off, 32);
    acc[k] = v;  // lane 0 of each wave holds the wave sum
  }
  const int wave = tid >> 5;
  const int lane = tid & 31;
  if (lane == 0) {
#pragma unroll
    for (int k = 0; k < NBINS; ++k) ssum[wave][k] = acc[k];
  }
  __syncthreads();
  if (tid < NBINS) {
    float s = 0.0f;
#pragma unroll
    for (int wv = 0; wv < THREADS / 32; ++wv) s += ssum[wv][tid];
    partial[((int64_t)plane * NCHUNKS + chunk) * NBINS + tid] = s;
  }
}

// ---------------------------------------------------------------------------
// Pass 2: fold the NCHUNKS partials per (plane,bin) into one_d (mean).
// Fixed summation order => bitwise deterministic across replays.
// ---------------------------------------------------------------------------
__global__ void hist_reduce(const float* __restrict__ partial,
                            float* __restrict__ one_d, int n) {
  const int t = blockIdx.x * blockDim.x + threadIdx.x;
  if (t >= n) return;
  const int plane = t / NBINS;
  const int k = t - plane * NBINS;
  float s = 0.0f;
#pragma unroll 8
  for (int ch = 0; ch < NCHUNKS; ++ch)
    s += partial[((int64_t)plane * NCHUNKS + ch) * NBINS + k];
  one_d[t] = s * (1.0f / (float)HWSZ);  // exact: /2^18
}

// ---------------------------------------------------------------------------
extern "C" void kernel_launch(void* const* d_in, const int* in_sizes, int n_in,
                              void* d_out, int out_size, void* d_ws, size_t ws_size,
                              hipStream_t stream) {
  (void)n_in; (void)out_size; (void)ws_size;
  const float* img     = (const float*)d_in[0];  // [B,C,H,W] f32
  const float* centers = (const float*)d_in[1];  // [8] f32
  const float* width   = (const float*)d_in[2];  // scalar f32

  const int planes = in_sizes[0] / HWSZ;         // B*C = 48

  float* one_d   = (float*)d_out;                        // [planes*NBINS]
  float* two_d   = one_d + (size_t)planes * NBINS;       // [planes*NBINS*HWSZ]
  float* partial = (float*)d_ws;                         // planes*NCHUNKS*NBINS f32 (96 KB)

  dim3 grid(NCHUNKS, planes);
  hist_main<<<grid, THREADS, 0, stream>>>(img, centers, width, two_d, partial);

  const int n = planes * NBINS;                  // 384
  hist_reduce<<<(n + 255) / 256, 256, 0, stream>>>(partial, one_d, n);
}